// Renderer_56805237457239
// MI455X (gfx1250) — compile-verified
//
#include <hip/hip_runtime.h>
#include <math.h>

typedef __attribute__((ext_vector_type(2))) float v2f;
typedef __attribute__((ext_vector_type(8))) float v8f;

#define B_   2
#define P_   300
#define F_   512
#define TH_  512
#define TW_  512
#define H_   128
#define W_   128
#define DELTA_ 7000.0f
#define REC  32   // floats per face record (128 bytes)

// Record layout (floats):
// [0..3]  w0 coeffs (a,b,c,0)   [4..7]  w1 coeffs   [8..11] w2 coeffs
// [12..15] z-plane coeffs       [16..21] x0,y0,x1,y1,x2,y2
// [22..27] u0,v0,u1,v1,u2,v2    [28] front flag     [29..31] pad

__global__ void face_setup(const float* __restrict__ pts,
                           const int*   __restrict__ faces,
                           const float* __restrict__ uv,
                           float*       __restrict__ ws)
{
    int idx = blockIdx.x * blockDim.x + threadIdx.x;
    if (idx >= B_ * F_) return;
    int b = idx / F_, f = idx % F_;
    int i0 = faces[f*3+0], i1 = faces[f*3+1], i2 = faces[f*3+2];
    const float* pb = pts + b*P_*3;
    float x0=pb[i0*3+0], y0=pb[i0*3+1], z0=pb[i0*3+2];
    float x1=pb[i1*3+0], y1=pb[i1*3+1], z1=pb[i1*3+2];
    float x2=pb[i2*3+0], y2=pb[i2*3+1], z2=pb[i2*3+2];
    float area = (x1-x0)*(y2-y0) - (y1-y0)*(x2-x0);
    bool front = area > 1e-8f;
    float inv = 1.0f / (front ? area : 1.0f);
    // edge(a->b, p) = (bx-ax)(py-ay) - (by-ay)(px-ax)
    float a0 = -(y2-y1)*inv, b0 = (x2-x1)*inv, c0 = ((y2-y1)*x1 - (x2-x1)*y1)*inv; // edge(v1,v2)
    float a1 = -(y0-y2)*inv, b1 = (x0-x2)*inv, c1 = ((y0-y2)*x2 - (x0-x2)*y2)*inv; // edge(v2,v0)
    float a2 = -(y1-y0)*inv, b2 = (x1-x0)*inv, c2 = ((y1-y0)*x0 - (x1-x0)*y0)*inv; // edge(v0,v1)
    // z-plane coeffs (before clobbering c0 for backfaces)
    float zx = a0*z0 + a1*z1 + a2*z2;
    float zy = b0*z0 + b1*z1 + b2*z2;
    float zc = c0*z0 + c1*z1 + c2*z2;
    if (!front) c0 = -1e38f;   // forces w0<0 -> inside==false for backfaces
    float* r = ws + idx*REC;
    r[0]=a0;  r[1]=b0;  r[2]=c0;  r[3]=0.f;
    r[4]=a1;  r[5]=b1;  r[6]=c1;  r[7]=0.f;
    r[8]=a2;  r[9]=b2;  r[10]=c2; r[11]=0.f;
    r[12]=zx; r[13]=zy; r[14]=zc; r[15]=0.f;
    r[16]=x0; r[17]=y0; r[18]=x1; r[19]=y1; r[20]=x2; r[21]=y2;
    const float* ub = uv + b*P_*2;
    r[22]=ub[i0*2+0]; r[23]=ub[i0*2+1];
    r[24]=ub[i1*2+0]; r[25]=ub[i1*2+1];
    r[26]=ub[i2*2+0]; r[27]=ub[i2*2+1];
    r[28]= front ? 1.f : 0.f;
    r[29]=0.f; r[30]=0.f; r[31]=0.f;
}

static __device__ __forceinline__ float seg_d2(float ax, float ay, float bx, float by,
                                               float px, float py)
{
    float abx = bx - ax, aby = by - ay;
    float apx = px - ax, apy = py - ay;
    float denom = abx*abx + aby*aby + 1e-12f;
    // v_rcp_f32 (1 ULP) instead of IEEE div expansion: t is clamped to [0,1]
    // and only feeds exp(-7000*d2); 1-ULP error here is ~1e-5 relative in p.
    float t = (apx*abx + apy*aby) * __builtin_amdgcn_rcpf(denom);
    t = fminf(fmaxf(t, 0.f), 1.f);
    float dx = apx - t*abx, dy = apy - t*aby;
    return dx*dx + dy*dy;
}

__global__ __launch_bounds__(128)
void render_kernel(const float* __restrict__ ws,
                   const float* __restrict__ tex,
                   float*       __restrict__ out)
{
    __shared__ float sf[F_*REC];          // 64 KB face table
    int tid  = threadIdx.x;
    int blk  = blockIdx.x;
    int b    = blk / 256;                 // 256 tiles per batch
    int tile = blk % 256;
    int tx   = tile % 8;                  // 8 tiles of 16px across
    int ty   = tile / 8;                  // 32 tiles of 4 rows down

    // ---- stage full face table into LDS with CDNA5 async global->LDS DMA ----
    // Each thread issues 32 async b128 copies (bypass VGPRs, tracked by ASYNCcnt).
    {
        const float4* src = (const float4*)(ws + (size_t)b*F_*REC);
        // Make the LDS array escape into the asm so its "memory" clobber is
        // known to alias sf[]; otherwise AA proves sf is never written and
        // deletes every LDS read (observed in round 2).
        float* sf_escape = sf;
        asm volatile("" : "+v"(sf_escape) :: "memory");
        #pragma unroll
        for (int j = 0; j < (F_*REC/4)/128; ++j) {
            int idx = j*128 + tid;
            unsigned lds_off = (unsigned)(idx * 16);   // sf[] starts at LDS offset 0
            const float4* gaddr = src + idx;
            asm volatile("global_load_async_to_lds_b128 %0, %1, off"
                         :: "v"(lds_off), "v"(gaddr) : "memory");
        }
        asm volatile("s_wait_asynccnt 0" ::: "memory");
    }
    __syncthreads();

    int  lane = tid & 31;
    int  wave = tid >> 5;
    int  n    = lane & 15;
    bool lo   = lane < 16;
    int  x    = tx*16 + n;
    int  y    = ty*4  + wave;
    float px  = (x + 0.5f) * (2.0f/W_) - 1.0f;
    float py  = 1.0f - (y + 0.5f) * (2.0f/H_);

    // B matrix (4x16): K rows {px, py, 1, 0}, column N = lane%16.
    // Lanes 0-15 carry K=(0,1), lanes 16-31 carry K=(2,3).
    v2f bmat;
    bmat.x = lo ? px : 1.0f;
    bmat.y = lo ? py : 0.0f;
    v8f cz = {0.f,0.f,0.f,0.f,0.f,0.f,0.f,0.f};

    float bestz = -1e30f;
    int   bidx  = -1;
    float bw0 = 0.f, bw1 = 0.f, bw2 = 0.f;
    float lm  = 0.f;

    const int fhalf = lo ? 0 : 8;   // D layout: VGPR m holds face m + 8*(lane>=16)
    const int koff  = lo ? 0 : 2;   // A layout: float offset of this lane's K pair

    for (int c = 0; c < F_/16; ++c) {
        const float* arec = sf + (c*16 + (lane & 15))*REC;
        v2f a0 = *(const v2f*)(arec + 0  + koff);
        v2f a1 = *(const v2f*)(arec + 4  + koff);
        v2f a2 = *(const v2f*)(arec + 8  + koff);
        v2f az = *(const v2f*)(arec + 12 + koff);
        // D[face, pixel] = coeffs x {px,py,1,0}  — 16 faces x 16 pixels per op
        v8f dw0 = __builtin_amdgcn_wmma_f32_16x16x4_f32(false, a0, false, bmat, (short)0, cz, false, false);
        v8f dw1 = __builtin_amdgcn_wmma_f32_16x16x4_f32(false, a1, false, bmat, (short)0, cz, false, false);
        v8f dw2 = __builtin_amdgcn_wmma_f32_16x16x4_f32(false, a2, false, bmat, (short)0, cz, false, false);
        v8f dz  = __builtin_amdgcn_wmma_f32_16x16x4_f32(false, az, false, bmat, (short)0, cz, false, false);
        #pragma unroll
        for (int m = 0; m < 8; ++m) {
            int fl = c*16 + m + fhalf;
            const float* r = sf + fl*REC;
            float w0 = dw0[m], w1 = dw1[m], w2 = dw2[m], z = dz[m];
            bool inside = (w0 >= 0.f) & (w1 >= 0.f) & (w2 >= 0.f);
            // branchless argmax update (v_cndmask instead of exec save/restore)
            bool upd = inside & (z > bestz);
            bestz = upd ? z  : bestz;
            bidx  = upd ? fl : bidx;
            bw0   = upd ? w0 : bw0;
            bw1   = upd ? w1 : bw1;
            bw2   = upd ? w2 : bw2;
            float d2 = fminf(fminf(seg_d2(r[16],r[17],r[18],r[19],px,py),
                                   seg_d2(r[18],r[19],r[20],r[21],px,py)),
                             seg_d2(r[20],r[21],r[16],r[17],px,py));
            d2 = inside ? 0.f : d2;
            float p = r[28] * __expf(-DELTA_ * d2);    // r[28]=front flag; native v_exp
            p = fminf(p, 1.0f - 1e-7f);
            lm += log1pf(-p);
        }
    }

    // ---- merge the two lane halves (each pixel is split across lane l and l+16) ----
    float z_o  = __shfl_xor(bestz, 16, 32);
    int   i_o  = __shfl_xor(bidx,  16, 32);
    float w0_o = __shfl_xor(bw0,   16, 32);
    float w1_o = __shfl_xor(bw1,   16, 32);
    float w2_o = __shfl_xor(bw2,   16, 32);
    float lm_t = lm + __shfl_xor(lm, 16, 32);
    bool take = (z_o > bestz) ||
                ((z_o == bestz) && (i_o >= 0) && (bidx < 0 || i_o < bidx));
    if (take) { bestz = z_o; bidx = i_o; bw0 = w0_o; bw1 = w1_o; bw2 = w2_o; }

    if (lo) {
        float hit = (bidx >= 0) ? 1.f : 0.f;
        int fb = (bidx >= 0) ? bidx : 0;
        const float* r = sf + fb*REC;
        float u    = (bw0*r[22] + bw1*r[24] + bw2*r[26]) * hit;
        float v    = (bw0*r[23] + bw1*r[25] + bw2*r[27]) * hit;
        float mask = (bw0 + bw1 + bw2) * hit;
        // fragment shader: bilinear sample
        float uc = fminf(fmaxf(u, 0.f), 1.f);
        float vc = fminf(fmaxf(v, 0.f), 1.f);
        float fx = uc * (float)(TW_-1);
        float fy = (1.0f - vc) * (float)(TH_-1);
        int x0i = (int)floorf(fx);
        int y0i = (int)floorf(fy);
        int x1i = min(x0i + 1, TW_-1);
        int y1i = min(y0i + 1, TH_-1);
        float wx = fx - (float)x0i, wy = fy - (float)y0i;
        const float* tb = tex + (size_t)b*3*TH_*TW_;
        int pix = (b*H_ + y)*W_ + x;
        #pragma unroll
        for (int ch = 0; ch < 3; ++ch) {
            const float* tc = tb + (size_t)ch*TH_*TW_;
            float c00 = tc[y0i*TW_ + x0i];
            float c01 = tc[y0i*TW_ + x1i];
            float c10 = tc[y1i*TW_ + x0i];
            float c11 = tc[y1i*TW_ + x1i];
            float col = c00*(1.f-wx)*(1.f-wy) + c01*wx*(1.f-wy)
                      + c10*(1.f-wx)*wy       + c11*wx*wy;
            out[pix*3 + ch] = col * mask;
        }
        out[B_*H_*W_*3 + pix] = 1.0f - expf(lm_t);
    }
}

extern "C" void kernel_launch(void* const* d_in, const int* in_sizes, int n_in,
                              void* d_out, int out_size, void* d_ws, size_t ws_size,
                              hipStream_t stream)
{
    const float* pts   = (const float*)d_in[0];
    const int*   faces = (const int*)  d_in[1];
    const float* uv    = (const float*)d_in[2];
    const float* tex   = (const float*)d_in[3];
    float* ws = (float*)d_ws;                       // B_*F_*REC*4 = 128 KB scratch

    face_setup<<<(B_*F_ + 255)/256, 256, 0, stream>>>(pts, faces, uv, ws);
    render_kernel<<<B_ * (H_/4) * (W_/16), 128, 0, stream>>>(ws, tex, (float*)d_out);
}